// SSIM_60155311947837
// MI455X (gfx1250) — compile-verified
//
#include <hip/hip_runtime.h>

// SSIM map over (32,3,512,512) fp32. Bandwidth-bound stencil:
// ~302 MB traffic -> ~13us floor at 23.3 TB/s. Strategy: async
// global->LDS tile staging (CDNA5 ASYNCcnt path), register sliding-window
// separable 3x3 box filters, one coalesced store.

#define IMG_H 512
#define IMG_W 512
#define TILE 64
#define HALO (TILE + 2)        // 66
#define LDS_STRIDE (HALO + 2)  // 68 words, pad to decouple rows
#define ROWS_PER_T 16          // 4 thread-rows * 16 = 64 rows
#define C1 0.0001f
#define C2 0.0009f

// ---- CDNA5 async global->LDS probes -------------------------------------
#if defined(__has_builtin)
#if __has_builtin(__builtin_amdgcn_global_load_async_to_lds_b32)
#define USE_ASYNC_LDS 1
#pragma message("CDNA5: using __builtin_amdgcn_global_load_async_to_lds_b32")
#endif
#if __has_builtin(__builtin_amdgcn_s_wait_asynccnt)
#define HAVE_WAIT_ASYNC_BUILTIN 1
#pragma message("CDNA5: using __builtin_amdgcn_s_wait_asynccnt")
#endif
#endif
#ifndef USE_ASYNC_LDS
#pragma message("CDNA5: async-to-LDS builtin NOT found, using sync fallback")
#endif

typedef __attribute__((address_space(1))) int gint_t;  // global
typedef __attribute__((address_space(3))) int lint_t;  // LDS

__device__ __forceinline__ void async_copy_b32(const float* __restrict__ gsrc,
                                               float* ldst) {
#ifdef USE_ASYNC_LDS
  // Builtin params are int* (as1) / int* (as3): strip const + retype in
  // generic space, then address-space cast.
  gint_t* g = (gint_t*)((int*)(const int*)gsrc);
  lint_t* l = (lint_t*)((int*)ldst);
  __builtin_amdgcn_global_load_async_to_lds_b32(g, l, /*imm offset=*/0,
                                                /*cpol=*/0);
#else
  *ldst = *gsrc;
#endif
}

__device__ __forceinline__ void wait_async_lds() {
#ifdef USE_ASYNC_LDS
#ifdef HAVE_WAIT_ASYNC_BUILTIN
  __builtin_amdgcn_s_wait_asynccnt(0);
#else
  asm volatile("s_wait_asynccnt 0" ::: "memory");
#endif
#endif
}

// jnp.pad mode='reflect' (no edge repeat): -1 -> 1, 512 -> 510
__device__ __forceinline__ int reflect_idx(int i) {
  if (i < 0) i = -i;
  if (i >= IMG_H) i = 2 * IMG_H - 2 - i;
  return i;
}

__global__ __launch_bounds__(256) void ssim_map_kernel(
    const float* __restrict__ x, const float* __restrict__ y,
    float* __restrict__ out) {
  __shared__ float ldsX[HALO * LDS_STRIDE];
  __shared__ float ldsY[HALO * LDS_STRIDE];

  const int plane = blockIdx.z;
  const int tileC0 = blockIdx.x * TILE;
  const int tileR0 = blockIdx.y * TILE;
  const size_t base = (size_t)plane * (IMG_H * IMG_W);

  const int tid = threadIdx.y * blockDim.x + threadIdx.x;  // 0..255

  // ---- Stage 66x66 halo of x and y into LDS via async copies ----
  for (int idx = tid; idx < HALO * HALO; idx += 256) {
    const int hr = idx / HALO;
    const int hc = idx - hr * HALO;
    const int gr = reflect_idx(tileR0 - 1 + hr);
    const int gc = reflect_idx(tileC0 - 1 + hc);
    const size_t goff = base + (size_t)gr * IMG_W + gc;
    const int loff = hr * LDS_STRIDE + hc;
    async_copy_b32(x + goff, &ldsX[loff]);
    async_copy_b32(y + goff, &ldsY[loff]);
  }
  wait_async_lds();
  __syncthreads();

  // ---- Sliding 3-row window of horizontal sums (registers) ----
  const int tx = threadIdx.x;                    // output column 0..63
  const int r_start = threadIdx.y * ROWS_PER_T;  // first output row

  float wsx[3], wsy[3], wsxx[3], wsyy[3], wsxy[3];

  auto hsum = [&](int hr, int k) {
    const float* px = &ldsX[hr * LDS_STRIDE + tx];
    const float* py = &ldsY[hr * LDS_STRIDE + tx];
    const float x0 = px[0], x1 = px[1], x2 = px[2];
    const float y0 = py[0], y1 = py[1], y2 = py[2];
    wsx[k] = x0 + x1 + x2;
    wsy[k] = y0 + y1 + y2;
    wsxx[k] = x0 * x0 + x1 * x1 + x2 * x2;
    wsyy[k] = y0 * y0 + y1 * y1 + y2 * y2;
    wsxy[k] = x0 * y0 + x1 * y1 + x2 * y2;
  };

  // Output row r uses halo rows r, r+1, r+2. Preload first two.
  hsum(r_start + 0, 0);
  hsum(r_start + 1, 1);

  const float inv9 = 1.0f / 9.0f;
#pragma unroll
  for (int i = 0; i < ROWS_PER_T; ++i) {
    hsum(r_start + i + 2, (i + 2) % 3);

    const float Sx = wsx[0] + wsx[1] + wsx[2];
    const float Sy = wsy[0] + wsy[1] + wsy[2];
    const float Sxx = wsxx[0] + wsxx[1] + wsxx[2];
    const float Syy = wsyy[0] + wsyy[1] + wsyy[2];
    const float Sxy = wsxy[0] + wsxy[1] + wsxy[2];

    const float mux = Sx * inv9;
    const float muy = Sy * inv9;
    const float sigx = Sxx * inv9 - mux * mux;
    const float sigy = Syy * inv9 - muy * muy;
    const float sigxy = Sxy * inv9 - mux * muy;

    const float n = (2.0f * mux * muy + C1) * (2.0f * sigxy + C2);
    const float d = (mux * mux + muy * muy + C1) * (sigx + sigy + C2);
    float v = (1.0f - n / d) * 0.5f;
    v = fminf(fmaxf(v, 0.0f), 1.0f);

    const int r = r_start + i;
    out[base + (size_t)(tileR0 + r) * IMG_W + (tileC0 + tx)] = v;
  }
}

extern "C" void kernel_launch(void* const* d_in, const int* in_sizes, int n_in,
                              void* d_out, int out_size, void* d_ws,
                              size_t ws_size, hipStream_t stream) {
  const float* x = (const float*)d_in[0];
  const float* y = (const float*)d_in[1];
  float* out = (float*)d_out;

  const int planes = in_sizes[0] / (IMG_H * IMG_W);  // 32*3 = 96
  dim3 grid(IMG_W / TILE, IMG_H / TILE, planes);
  dim3 block(TILE, 4, 1);
  ssim_map_kernel<<<grid, block, 0, stream>>>(x, y, out);
}